// EdgeFeatureExpansion_55430847922653
// MI455X (gfx1250) — compile-verified
//
#include <hip/hip_runtime.h>
#include <hip/hip_bf16.h>
#include <stdint.h>

#define EPSF 1e-5f

typedef float v4f __attribute__((ext_vector_type(4)));
typedef float v2f __attribute__((ext_vector_type(2)));
typedef unsigned int v4u __attribute__((ext_vector_type(4)));
typedef int v8i __attribute__((ext_vector_type(8)));
typedef int v4i __attribute__((ext_vector_type(4)));

#define TILE_EDGES 64
#define WAVES_PER_BLOCK 8
#define THREADS (WAVES_PER_BLOCK * 32)
#define TILE_ELEMS (TILE_EDGES * 8) /* 512 floats of edge_attr per tile */

#if __has_builtin(__builtin_amdgcn_tensor_load_to_lds) && __has_builtin(__builtin_amdgcn_s_wait_tensorcnt)
#define HAVE_TDM 1
#else
#define HAVE_TDM 0
#endif

__device__ __forceinline__ uint32_t rfl(uint32_t x) {
  return (uint32_t)__builtin_amdgcn_readfirstlane((int)x);
}

// Intra-quad XOR-add on the VALU pipe (DPP quad_perm), keeps the DS pipe free
// for the TDM-staged attr reads. ctrl 0xB1 = [1,0,3,2] (xor 1), 0x4E = [2,3,0,1] (xor 2).
template <int CTRL>
__device__ __forceinline__ float quad_xor_add(float x) {
  int xi = __float_as_int(x);
  int yi = __builtin_amdgcn_update_dpp(0, xi, CTRL, 0xF, 0xF, true);
  return x + __int_as_float(yi);
}

#if HAVE_TDM
// 1-D TDM copy: nelems f32 from gptr -> LDS byte offset lds_off.
// Descriptor per CDNA5 ISA 8.3/8.4: count=1, type=2, data_size=4B,
// tensor_dim0=nelems (OOB reads return 0), tile_dim0=TILE_ELEMS, dims1..=1.
// 6-arg builtin on this toolchain: (g0:v4u, g1:v8i, g2:v4i, g3:v4i, g4:v8i, cpol).
__device__ __forceinline__ void tdm_load_attr(uint32_t lds_off, const float* gptr,
                                              uint32_t nelems) {
  uint64_t ga = (uint64_t)(uintptr_t)gptr;
  v4u g0;
  g0[0] = rfl(1u);                                            // count=1
  g0[1] = rfl(lds_off);                                       // lds_addr (bytes)
  g0[2] = rfl((uint32_t)ga);                                  // global_addr[31:0]
  g0[3] = rfl(((uint32_t)(ga >> 32) & 0x01FFFFFFu) | (2u << 30)); // ga[56:32] | type=2
  v8i g1;
  g1[0] = (int)rfl(2u << 16);                                 // data_size=4B, mask=0
  g1[1] = (int)rfl((nelems & 0xFFFFu) << 16);                 // tensor_dim0[15:0]
  g1[2] = (int)rfl(((nelems >> 16) & 0xFFFFu) | (1u << 16));  // td0[31:16] | tensor_dim1=1
  g1[3] = (int)rfl(((uint32_t)TILE_ELEMS) << 16);             // tile_dim0
  g1[4] = (int)rfl(1u);                                       // tile_dim1=1, tile_dim2=0
  g1[5] = (int)rfl(nelems);                                   // tensor_dim0_stride lo
  g1[6] = 0;
  g1[7] = 0;
  v4i z4 = (v4i)0;                                            // groups 2/3 unused (2-D)
  v8i z8 = (v8i)0;
  __builtin_amdgcn_tensor_load_to_lds(g0, g1, z4, z4, z8, 0);
}
#endif

__global__ __launch_bounds__(THREADS) void edge_expand_kernel(
    const float* __restrict__ node_feat,   // [N,16]
    const float* __restrict__ edge_attr,   // [M,8]
    const int* __restrict__ edge_index,    // [2,M]
    float* __restrict__ out,               // [M,100]
    const long long M, const int nTiles, const int totalWaves) {
#if HAVE_TDM
  __shared__ float attrLds[WAVES_PER_BLOCK][2][TILE_ELEMS]; // 32 KB
#endif
  const int lane = threadIdx.x & 31;
  const int wib = threadIdx.x >> 5;  // wave in block (uniform per wave)
  const int quad = lane >> 2;        // 0..7: edge within 8-edge step
  const int ql = lane & 3;           // 0..3: float4 slot within edge
  const int t0 = (int)blockIdx.x * WAVES_PER_BLOCK + wib;
  if (t0 >= nTiles) return;

#if HAVE_TDM
  const uint32_t ldsBase0 = rfl((uint32_t)(uintptr_t)&attrLds[wib][0][0]);
  const uint32_t ldsBase1 = rfl((uint32_t)(uintptr_t)&attrLds[wib][1][0]);
  { // prime the pipeline: tile t0 -> buf0
    long long eb = (long long)t0 * TILE_EDGES;
    long long remLL = (M - eb) * 8;
    uint32_t rem = (remLL > TILE_ELEMS) ? (uint32_t)TILE_ELEMS : (uint32_t)remLL;
    tdm_load_attr(ldsBase0, edge_attr + eb * 8, rem);
  }
#endif

  int buf = 0;
  for (int t = t0; t < nTiles; t += totalWaves, buf ^= 1) {
    const int tn = t + totalWaves;
#if HAVE_TDM
    if (tn < nTiles) { // prefetch next tile into the other buffer, keep 1 in flight
      long long eb = (long long)tn * TILE_EDGES;
      long long remLL = (M - eb) * 8;
      uint32_t rem = (remLL > TILE_ELEMS) ? (uint32_t)TILE_ELEMS : (uint32_t)remLL;
      tdm_load_attr(buf ? ldsBase0 : ldsBase1, edge_attr + eb * 8, rem);
      __builtin_amdgcn_s_wait_tensorcnt(1);
    } else {
      __builtin_amdgcn_s_wait_tensorcnt(0);
    }
    const float* abuf = &attrLds[wib][buf][0];
#endif
    const long long eBase = (long long)t * TILE_EDGES;
#pragma unroll
    for (int s = 0; s < TILE_EDGES / 8; ++s) {
      const long long e = eBase + s * 8 + quad;
      const bool valid = (e < M);
      const long long es = valid ? e : 0;

      // gather indices (streamed once -> non-temporal)
      const int idst = __builtin_nontemporal_load(edge_index + es);      // row 0
      const int isrc = __builtin_nontemporal_load(edge_index + M + es);  // row 1

      // node_feat gathers: 6.4 MB table, L2-resident -> regular (RT) b128 loads
      const v4f sf = *(const v4f*)(node_feat + (size_t)(uint32_t)isrc * 16 + ql * 4);
      const v4f df = *(const v4f*)(node_feat + (size_t)(uint32_t)idst * 16 + ql * 4);
      const v4f dv = sf - df;

      // ||diff||^2: 4 partial elems per lane, quad reduction via DPP (VALU pipe)
      float part = dv[0] * dv[0] + dv[1] * dv[1] + dv[2] * dv[2] + dv[3] * dv[3];
      float p2 = quad_xor_add<0xB1>(part);   // quad_perm [1,0,3,2]
      const float sum = quad_xor_add<0x4E>(p2);  // quad_perm [2,3,0,1]

      const float norm = sqrtf(sum);
      const float ninv = 1.0f / (norm + EPSF);
      const v4f unit = dv * ninv;
      const float n2 = norm * norm;
      const float ninv2 = ninv * ninv;

#if HAVE_TDM
      const v2f a01 = *(const v2f*)(abuf + (s * 8 + quad) * 8 + ql * 2);
#else
      const v2f a01 = __builtin_nontemporal_load(
          (const v2f*)(edge_attr + (size_t)es * 8 + ql * 2));
#endif
      v2f ainv, asq, ainv2;
      ainv[0] = 1.0f / (a01[0] + EPSF);
      ainv[1] = 1.0f / (a01[1] + EPSF);
      asq[0] = a01[0] * a01[0];
      asq[1] = a01[1] * a01[1];
      ainv2[0] = ainv[0] * ainv[0];
      ainv2[1] = ainv[1] * ainv[1];

      if (valid) {
        // 1.28 GB output stream -> non-temporal so L2 keeps node_feat hot
        float* o = out + (size_t)e * 100;  // 400 B rows, 16 B aligned
        __builtin_nontemporal_store(a01, (v2f*)(o + 2 * ql));          // [0,8)   attr
        __builtin_nontemporal_store(sf, (v4f*)(o + 8 + 4 * ql));       // [8,24)  src
        __builtin_nontemporal_store(df, (v4f*)(o + 24 + 4 * ql));      // [24,40) dst
        __builtin_nontemporal_store(dv, (v4f*)(o + 40 + 4 * ql));      // [40,56) diff
        __builtin_nontemporal_store(unit, (v4f*)(o + 56 + 4 * ql));    // [56,72) unit
        const float sc = (ql == 0) ? norm : (ql == 1) ? ninv : (ql == 2) ? n2 : ninv2;
        __builtin_nontemporal_store(sc, o + 72 + ql);                  // [72,76) scalars
        __builtin_nontemporal_store(ainv, (v2f*)(o + 76 + 2 * ql));    // [76,84) attr_inv
        __builtin_nontemporal_store(asq, (v2f*)(o + 84 + 2 * ql));     // [84,92) attr_sq
        __builtin_nontemporal_store(ainv2, (v2f*)(o + 92 + 2 * ql));   // [92,100) attr_inv_sq
      }
    }
  }
}

extern "C" void kernel_launch(void* const* d_in, const int* in_sizes, int n_in,
                              void* d_out, int out_size, void* d_ws, size_t ws_size,
                              hipStream_t stream) {
  const float* node_feat = (const float*)d_in[0];
  const float* edge_attr = (const float*)d_in[1];
  const int* edge_index = (const int*)d_in[2];
  float* out = (float*)d_out;

  const long long M = (long long)in_sizes[1] / 8;  // edge_attr is [M,8]
  const int nTiles = (int)((M + TILE_EDGES - 1) / TILE_EDGES);

  int blocks = 640;  // persistent-ish: 5120 waves, ~10 tiles each at M=3.2M
  if (blocks * WAVES_PER_BLOCK > nTiles)
    blocks = (nTiles + WAVES_PER_BLOCK - 1) / WAVES_PER_BLOCK;
  const int totalWaves = blocks * WAVES_PER_BLOCK;

  edge_expand_kernel<<<blocks, THREADS, 0, stream>>>(node_feat, edge_attr, edge_index,
                                                     out, M, nTiles, totalWaves);
}